// MetapathAggrLayer_1580547970287
// MI455X (gfx1250) — compile-verified
//
#include <hip/hip_runtime.h>

// CDNA5 / gfx1250: wave32, WMMA (not MFMA).
typedef __attribute__((ext_vector_type(2))) float v2f;
typedef __attribute__((ext_vector_type(8))) float v8f;

#define ALPHA   0.2f
#define FDIM    64
#define NMETA   4
#define CHUNKS  16          // FDIM / 4  (K per V_WMMA_F32_16X16X4_F32)
#define BLOCK   256
#define WAVES   (BLOCK / 32)
#define NPW     16          // nodes per wave (16x16 WMMA tile rows)

__global__ __launch_bounds__(BLOCK)
void metapath_aggr_kernel(const float* __restrict__ in,   // [NMETA, N, FDIM]
                          const float* __restrict__ a,    // [FDIM]
                          float* __restrict__ out,        // [N, FDIM]
                          int N)
{
    __shared__ float a_lds[FDIM];
    __shared__ float e_lds[WAVES][NMETA][NPW];

    const int tid = threadIdx.x;
    if (tid < FDIM) a_lds[tid] = a[tid];
    __syncthreads();

    const int lane  = tid & 31;
    const int w     = tid >> 5;
    const int hi    = lane >> 4;      // 0 -> features {4c,4c+1}, 1 -> {4c+2,4c+3}
    const int nodeL = lane & 15;      // node within the 16-node tile
    const long long node0 = ((long long)blockIdx.x * WAVES + w) * NPW;
    if (node0 >= N) return;

    // ---- Load X tiles for all 4 metapaths directly in WMMA A-fragment layout.
    // A (16x4 f32): v0 = K=0 (lanes 0-15) | K=2 (lanes 16-31); v1 = K=1 | K=3.
    v2f ax[NMETA][CHUNKS];
    #pragma unroll
    for (int m = 0; m < NMETA; ++m) {
        const float* src = in + (((long long)m * N) + node0 + nodeL) * FDIM + hi * 2;
        #pragma unroll
        for (int c = 0; c < CHUNKS; ++c)
            ax[m][c] = *(const v2f*)(src + c * 4);
    }

    // ---- Scores: E_m = X_m (16x64) * A_bcast (64x16) via fp32 WMMA chain.
    // B fragment: every column of B is `a`, so D columns come out replicated.
    v8f acc[NMETA];
    #pragma unroll
    for (int m = 0; m < NMETA; ++m) acc[m] = {};

    #pragma unroll
    for (int c = 0; c < CHUNKS; ++c) {
        v2f bf = *(const v2f*)&a_lds[c * 4 + hi * 2];
        #pragma unroll
        for (int m = 0; m < NMETA; ++m) {
            acc[m] = __builtin_amdgcn_wmma_f32_16x16x4_f32(
                         false, ax[m][c], false, bf,
                         (short)0, acc[m], false, false);
        }
    }

    // ---- D layout: VGPR r holds e[node r] (lanes 0-15) / e[node r+8] (lanes 16-31).
    // Lanes 0 and 16 scatter the 16 per-metapath scores into this wave's LDS slice.
    if (nodeL == 0) {
        #pragma unroll
        for (int m = 0; m < NMETA; ++m) {
            #pragma unroll
            for (int r = 0; r < 8; ++r)
                e_lds[w][m][hi * 8 + r] = acc[m][r];
        }
    }
    asm volatile("s_wait_dscnt 0x0" ::: "memory");   // same-wave LDS RAW fence

    // ---- LeakyReLU + softmax over the 4 metapaths for this lane's node.
    float e[NMETA];
    float emax = -3.402823466e38f;
    #pragma unroll
    for (int m = 0; m < NMETA; ++m) {
        float v = e_lds[w][m][nodeL];
        v = (v >= 0.0f) ? v : (ALPHA * v);
        e[m] = v;
        emax = fmaxf(emax, v);
    }
    float s = 0.0f;
    #pragma unroll
    for (int m = 0; m < NMETA; ++m) { e[m] = __expf(e[m] - emax); s += e[m]; }
    const float inv = 1.0f / s;
    #pragma unroll
    for (int m = 0; m < NMETA; ++m) e[m] *= inv;

    // ---- Weighted sum over metapaths using the X fragments still in VGPRs.
    float* dst = out + (node0 + nodeL) * FDIM + hi * 2;
    #pragma unroll
    for (int c = 0; c < CHUNKS; ++c) {
        v2f o = e[0] * ax[0][c];
        #pragma unroll
        for (int m = 1; m < NMETA; ++m) o += e[m] * ax[m][c];
        *(v2f*)(dst + c * 4) = o;
    }
}

extern "C" void kernel_launch(void* const* d_in, const int* in_sizes, int n_in,
                              void* d_out, int out_size, void* d_ws, size_t ws_size,
                              hipStream_t stream) {
    const float* in = (const float*)d_in[0];   // [4, N, 64] fp32
    const float* a  = (const float*)d_in[1];   // [64, 1]    fp32
    float* out = (float*)d_out;                // [N, 64]    fp32

    const int N = in_sizes[0] / (NMETA * FDIM);
    const int nodesPerBlock = WAVES * NPW;     // 128
    const int grid = (N + nodesPerBlock - 1) / nodesPerBlock;

    metapath_aggr_kernel<<<grid, BLOCK, 0, stream>>>(in, a, out, N);
}